// LMSolver_42296837931415
// MI455X (gfx1250) — compile-verified
//
#include <hip/hip_runtime.h>

typedef __attribute__((ext_vector_type(2))) float v2f;
typedef __attribute__((ext_vector_type(8))) float v8f;

#define NPTS      512
#define ROWS      1024            // 2*NPTS residual rows
#define RSTRIDE   8               // dwords per row in LDS
#define LDS_DW    (ROWS * RSTRIDE + 32)  // +pad for masked OOB-ish reads
#define N_ITER    10
#define Z_MIN     0.1f
#define EPS_GN    1e-5f
#define MIN_DIAG  1e-6f
#define MAX_DIAG  1e32f
#define MIN_RELD  1e-3f
#define INIT_RAD  30.0f
#define MAX_RAD   1e16f

// Row r lives at dword base(r) = 8*r + 4*((r>>3)&3)   (swizzle keeps both the
// b128 stores and the WMMA-layout reads bank-conflict-free).
__device__ __forceinline__ int row_base(int r) {
    return r * RSTRIDE + (((r >> 3) & 3) << 2);
}

template <bool JAC>
__device__ __forceinline__ void point_res_jac(
    float X, float Y, float Z, float cs, float sn,
    float tx, float ty, float tz, const float* Kc,
    float u2, float v2, float wu, float wv,
    float& rx, float& ry, float* jx, float* jy)
{
    float pcx = fmaf(cs, X, fmaf(sn, Z, tx));
    float pcy = Y + ty;
    float pcz = fmaf(-sn, X, fmaf(cs, Z, tz));
    float uh = Kc[0]*pcx + Kc[1]*pcy + Kc[2]*pcz;
    float vh = Kc[3]*pcx + Kc[4]*pcy + Kc[5]*pcz;
    float wh = Kc[6]*pcx + Kc[7]*pcy + Kc[8]*pcz;
    float zc  = fmaxf(wh, Z_MIN);
    float inv = 1.0f / zc;
    float u = uh * inv, v = vh * inv;
    rx = (u - u2) * wu;
    ry = (v - v2) * wv;
    if (JAC) {
        float gz = (wh > Z_MIN) ? 1.0f : 0.0f;
        float ug = u * gz, vg = v * gz;
        float wxi = wu * inv, wyi = wv * inv;
        // d(cam point)/dyaw = (ax, 0, az)
        float ax = fmaf(-sn, X,  cs * Z);
        float az = fmaf(-cs, X, -sn * Z);
        float duh3 = Kc[0]*ax + Kc[2]*az;
        float dvh3 = Kc[3]*ax + Kc[5]*az;
        float dwh3 = Kc[6]*ax + Kc[8]*az;
        jx[0] = wxi * (Kc[0] - ug * Kc[6]);
        jx[1] = wxi * (Kc[1] - ug * Kc[7]);
        jx[2] = wxi * (Kc[2] - ug * Kc[8]);
        jx[3] = wxi * (duh3  - ug * dwh3);
        jy[0] = wyi * (Kc[3] - vg * Kc[6]);
        jy[1] = wyi * (Kc[4] - vg * Kc[7]);
        jy[2] = wyi * (Kc[5] - vg * Kc[8]);
        jy[3] = wyi * (dvh3  - vg * dwh3);
    }
}

// Build the 5x5 normal system M = G^T G with G = [J | r] (1024x5) using
// V_WMMA_F32_16X16X4_F32. A and B operands share the same lane data because
// A[m][k] = G[r0+k][m] and B[k][n] = G[r0+k][n] land on identical VGPR slots.
__device__ void build_normal(const float* __restrict__ X3,
                             const float* __restrict__ X2,
                             const float* __restrict__ W2,
                             const float* Kc,
                             float tx, float ty, float tz, float yaw,
                             float* g, int lane,
                             float JT[4][4], float grad[4], float& rr)
{
    float sn = sinf(yaw), cs = cosf(yaw);

    // Phase 1: each lane computes 16 points, stages rows [jx0..3, rx] / [jy0..3, ry]
    #pragma unroll 2
    for (int i = 0; i < NPTS / 32; ++i) {
        int p = lane + 32 * i;
        float X = X3[3*p], Y = X3[3*p + 1], Z = X3[3*p + 2];
        float rx, ry, jx[4], jy[4];
        point_res_jac<true>(X, Y, Z, cs, sn, tx, ty, tz, Kc,
                            X2[2*p], X2[2*p + 1], W2[2*p], W2[2*p + 1],
                            rx, ry, jx, jy);
        int base = row_base(2 * p);          // rows 2p, 2p+1 share the swizzle
        float4 a = make_float4(jx[0], jx[1], jx[2], jx[3]);
        float4 b = make_float4(jy[0], jy[1], jy[2], jy[3]);
        *(float4*)(g + base)     = a;  g[base + 4]           = rx;
        *(float4*)(g + base + RSTRIDE) = b;  g[base + RSTRIDE + 4] = ry;
    }
    __syncthreads();

    // Phase 2: 256 WMMA steps, each reducing 4 rows (K=4).
    const int  cL    = lane & 15;
    const int  kHalf = (lane >> 4) << 1;          // 0 or 2
    const bool cv    = (cL < 5);
    const int  off0  = kHalf * RSTRIDE + cL;
    v8f acc = {0.f, 0.f, 0.f, 0.f, 0.f, 0.f, 0.f, 0.f};
    #pragma unroll 4
    for (int r0 = 0; r0 < ROWS; r0 += 4) {
        int base = row_base(r0);                  // valid for rows r0..r0+3
        float a0 = g[base + off0];
        float a1 = g[base + off0 + RSTRIDE];
        v2f ab;
        ab.x = cv ? a0 : 0.0f;                    // VGPR0: K = kHalf
        ab.y = cv ? a1 : 0.0f;                    // VGPR1: K = kHalf+1
        acc = __builtin_amdgcn_wmma_f32_16x16x4_f32(
                  false, ab, false, ab, (short)0, acc, false, false);
    }
    __syncthreads();

    // Extract 5x5 corner: D element [m][n] sits in VGPR m (m<8), lane n.
    JT[0][0] = __shfl(acc[0], 0, 32);
    JT[0][1] = JT[1][0] = __shfl(acc[0], 1, 32);
    JT[0][2] = JT[2][0] = __shfl(acc[0], 2, 32);
    JT[0][3] = JT[3][0] = __shfl(acc[0], 3, 32);
    grad[0]  = __shfl(acc[0], 4, 32);
    JT[1][1] = __shfl(acc[1], 1, 32);
    JT[1][2] = JT[2][1] = __shfl(acc[1], 2, 32);
    JT[1][3] = JT[3][1] = __shfl(acc[1], 3, 32);
    grad[1]  = __shfl(acc[1], 4, 32);
    JT[2][2] = __shfl(acc[2], 2, 32);
    JT[2][3] = JT[3][2] = __shfl(acc[2], 3, 32);
    grad[2]  = __shfl(acc[2], 4, 32);
    JT[3][3] = __shfl(acc[3], 3, 32);
    grad[3]  = __shfl(acc[3], 4, 32);
    rr       = __shfl(acc[4], 4, 32);             // = sum r^2 = 2*cost
}

__device__ float cost_at(const float* __restrict__ X3,
                         const float* __restrict__ X2,
                         const float* __restrict__ W2,
                         const float* Kc,
                         float tx, float ty, float tz, float yaw, int lane)
{
    float sn = sinf(yaw), cs = cosf(yaw);
    float loc = 0.0f;
    #pragma unroll 4
    for (int i = 0; i < NPTS / 32; ++i) {
        int p = lane + 32 * i;
        float X = X3[3*p], Y = X3[3*p + 1], Z = X3[3*p + 2];
        float rx, ry, jd[4];
        point_res_jac<false>(X, Y, Z, cs, sn, tx, ty, tz, Kc,
                             X2[2*p], X2[2*p + 1], W2[2*p], W2[2*p + 1],
                             rx, ry, jd, jd);
        loc = fmaf(rx, rx, fmaf(ry, ry, loc));
    }
    #pragma unroll
    for (int o = 16; o > 0; o >>= 1) loc += __shfl_xor(loc, o, 32);
    return 0.5f * loc;
}

__device__ __forceinline__ void solve4(float A[4][4], float b[4], float x[4])
{
    #pragma unroll
    for (int i = 0; i < 4; ++i) {
        float inv = 1.0f / A[i][i];
        #pragma unroll
        for (int j = i + 1; j < 4; ++j) {
            float f = A[j][i] * inv;
            #pragma unroll
            for (int k = i; k < 4; ++k) A[j][k] -= f * A[i][k];
            b[j] -= f * b[i];
        }
    }
    #pragma unroll
    for (int i = 3; i >= 0; --i) {
        float s = b[i];
        #pragma unroll
        for (int j = i + 1; j < 4; ++j) s -= A[i][j] * x[j];
        x[i] = s / A[i][i];
    }
}

__global__ __launch_bounds__(32)
void lm_solver_kernel(const float* __restrict__ x3d,
                      const float* __restrict__ x2d,
                      const float* __restrict__ w2d,
                      const float* __restrict__ pose_init,
                      const float* __restrict__ cam,
                      float* __restrict__ out, int B)
{
    __shared__ __align__(16) float g[LDS_DW];
    const int b    = blockIdx.x;
    const int lane = threadIdx.x;

    const float* X3 = x3d + (size_t)b * NPTS * 3;
    const float* X2 = x2d + (size_t)b * NPTS * 2;
    const float* W2 = w2d + (size_t)b * NPTS * 2;

    float Kc[9];
    #pragma unroll
    for (int i = 0; i < 9; ++i) Kc[i] = cam[(size_t)b * 9 + i];

    float px = pose_init[b*4 + 0], py = pose_init[b*4 + 1];
    float pz = pose_init[b*4 + 2], pw = pose_init[b*4 + 3];

    float radius = INIT_RAD, dec = 2.0f, cost_fin = 0.0f;
    float JT[4][4], grad[4], rr;

    for (int it = 0; it < N_ITER; ++it) {
        build_normal(X3, X2, W2, Kc, px, py, pz, pw, g, lane, JT, grad, rr);
        float cost_cur = 0.5f * rr;

        float A[4][4], bb[4], step[4];
        #pragma unroll
        for (int i = 0; i < 4; ++i) {
            #pragma unroll
            for (int j = 0; j < 4; ++j) A[i][j] = JT[i][j];
            float d = fminf(fmaxf(JT[i][i], MIN_DIAG), MAX_DIAG);
            A[i][i] = JT[i][i] + d / radius;
            bb[i]   = -grad[i];
        }
        solve4(A, bb, step);

        float nx = px + step[0], ny = py + step[1];
        float nz = pz + step[2], nw = pw + step[3];
        float cost_new = cost_at(X3, X2, W2, Kc, nx, ny, nz, nw, lane);

        float js0 = JT[0][0]*step[0] + JT[0][1]*step[1] + JT[0][2]*step[2] + JT[0][3]*step[3];
        float js1 = JT[1][0]*step[0] + JT[1][1]*step[1] + JT[1][2]*step[2] + JT[1][3]*step[3];
        float js2 = JT[2][0]*step[0] + JT[2][1]*step[1] + JT[2][2]*step[2] + JT[2][3]*step[3];
        float js3 = JT[3][0]*step[0] + JT[3][1]*step[1] + JT[3][2]*step[2] + JT[3][3]*step[3];
        float mcc = -(step[0]*grad[0] + step[1]*grad[1] + step[2]*grad[2] + step[3]*grad[3]
                      + 0.5f*(step[0]*js0 + step[1]*js1 + step[2]*js2 + step[3]*js3));

        float rel = (cost_cur - cost_new) / mcc;
        bool success = (rel >= MIN_RELD) && (mcc > 0.0f);

        float t  = 2.0f * rel - 1.0f;
        float dn = fmaxf(1.0f - t*t*t, 1.0f/3.0f);
        float rs = fminf(radius / dn, MAX_RAD);
        radius = success ? rs : radius / dec;
        dec    = success ? 2.0f : dec * 2.0f;

        px = success ? nx : px;  py = success ? ny : py;
        pz = success ? nz : pz;  pw = success ? nw : pw;
        cost_fin = success ? cost_new : cost_cur;
    }

    // Final Gauss-Newton step at the optimized pose.
    build_normal(X3, X2, W2, Kc, px, py, pz, pw, g, lane, JT, grad, rr);
    float A[4][4], bb[4], step[4];
    #pragma unroll
    for (int i = 0; i < 4; ++i) {
        #pragma unroll
        for (int j = 0; j < 4; ++j) A[i][j] = JT[i][j];
        A[i][i] += EPS_GN;
        bb[i] = -grad[i];
    }
    solve4(A, bb, step);

    if (lane == 0) {
        out[b*4 + 0] = px;  out[b*4 + 1] = py;
        out[b*4 + 2] = pz;  out[b*4 + 3] = pw;
        out[4*B + b] = cost_fin;
        float* pp = out + 5*B + b*4;
        pp[0] = px + step[0];  pp[1] = py + step[1];
        pp[2] = pz + step[2];  pp[3] = pw + step[3];
    }
}

extern "C" void kernel_launch(void* const* d_in, const int* in_sizes, int n_in,
                              void* d_out, int out_size, void* d_ws, size_t ws_size,
                              hipStream_t stream) {
    const float* x3d  = (const float*)d_in[0];
    const float* x2d  = (const float*)d_in[1];
    const float* w2d  = (const float*)d_in[2];
    const float* pose = (const float*)d_in[3];
    const float* cam  = (const float*)d_in[4];
    float* out = (float*)d_out;
    int B = in_sizes[3] / 4;
    lm_solver_kernel<<<dim3(B), dim3(32), 0, stream>>>(x3d, x2d, w2d, pose, cam, out, B);
}